// PRIMEAugmentation32_24567212933296
// MI455X (gfx1250) — compile-verified
//
#include <hip/hip_runtime.h>
#include <math.h>

typedef __attribute__((ext_vector_type(2))) float v2f;
typedef __attribute__((ext_vector_type(8))) float v8f;

#define NN 256      // image spatial size
#define CUTN 100    // modes per axis
#define KP 112      // padded mode dimension (multiple of 16)

// ---------------------------------------------------------------------------
// prep: S[x][i] = sin(pi * x/(N-1) * (i+1))  (zero-padded cols i>=100)
//       M[i][j] = scale * c[i][j] / r,  r=sqrt((i+1)^2+(j+1)^2), if r<100.5
// ---------------------------------------------------------------------------
__global__ void prep_kernel(const float* __restrict__ cu,
                            const float* __restrict__ cv,
                            float* __restrict__ S,
                            float* __restrict__ Mu,
                            float* __restrict__ Mv,
                            float scale)
{
    int idx = blockIdx.x * blockDim.x + threadIdx.x;
    const float PI = 3.14159265358979323846f;
    if (idx < NN * KP) {
        int x = idx / KP;
        int i = idx % KP;
        float v = 0.0f;
        if (i < CUTN) {
            float xf = (float)x / (float)(NN - 1);
            v = sinf(PI * xf * (float)(i + 1));
        }
        S[idx] = v;
    } else {
        int t = idx - NN * KP;
        if (t < 2 * KP * KP) {
            int f = t / (KP * KP);
            int r = t % (KP * KP);
            int i = r / KP;
            int j = r % KP;
            float val = 0.0f;
            if (i < CUTN && j < CUTN) {
                float fi = (float)(i + 1), fj = (float)(j + 1);
                float rad = sqrtf(fi * fi + fj * fj);
                if (rad < (float)CUTN + 0.5f) {
                    const float* c = f ? cv : cu;
                    val = scale * c[i * CUTN + j] / rad;
                }
            }
            (f ? Mv : Mu)[r] = val;
        }
    }
}

// ---------------------------------------------------------------------------
// GEMM 1: P = S (256 x 112) @ M (112 x 112), one wave per 16x16 tile.
// Uses V_WMMA_F32_16X16X4_F32.  A frag: lanes 0-15 hold K=k0..k0+1 in v.x/v.y,
// lanes 16-31 hold K=k0+2..k0+3.  B frag mirrors with N across lanes.
// ---------------------------------------------------------------------------
__global__ void gemm_SM_kernel(const float* __restrict__ S,
                               const float* __restrict__ Mu,
                               const float* __restrict__ Mv,
                               float* __restrict__ Pu,
                               float* __restrict__ Pv)
{
    int tile  = blockIdx.x;
    int field = tile / (16 * 7);
    tile      = tile % (16 * 7);
    int tm = tile / 7;           // 0..15 row tiles  (256/16)
    int tn = tile % 7;           // 0..6  col tiles  (112/16)
    const float* B = field ? Mv : Mu;
    float*       D = field ? Pv : Pu;

    int lane  = (int)(threadIdx.x & 31u);
    int lo    = lane & 15;                 // M (for A) / N (for B) index
    int khalf = (lane >> 4) << 1;          // 0 or 2
    int row0  = tm * 16, col0 = tn * 16;

    v8f acc = {};
    for (int k0 = 0; k0 < KP; k0 += 4) {
        v2f a, b;
        const float* arow = S + (size_t)(row0 + lo) * KP + k0 + khalf;
        a.x = arow[0];
        a.y = arow[1];
        b.x = B[(size_t)(k0 + khalf) * KP + col0 + lo];
        b.y = B[(size_t)(k0 + khalf + 1) * KP + col0 + lo];
        acc = __builtin_amdgcn_wmma_f32_16x16x4_f32(
                  false, a, false, b, (short)0, acc, false, false);
    }
#pragma unroll
    for (int r = 0; r < 8; ++r) {
        int m = r + ((lane >> 4) << 3);
        D[(size_t)(row0 + m) * KP + col0 + lo] = acc[r];
    }
}

// ---------------------------------------------------------------------------
// GEMM 2: U'[x,y] = sum_j P[x,j] * S[y,j]  (i.e. U' = P @ S^T, 256x256).
// B fragment loads S transposed by addressing: B[k][n] = S[col0+n][k].
// ---------------------------------------------------------------------------
__global__ void gemm_PSt_kernel(const float* __restrict__ S,
                                const float* __restrict__ Pu,
                                const float* __restrict__ Pv,
                                float* __restrict__ DX,
                                float* __restrict__ DY)
{
    int tile  = blockIdx.x;
    int field = tile / 256;
    tile      = tile % 256;
    int tm = tile / 16;          // 0..15 row tiles (x)
    int tn = tile % 16;          // 0..15 col tiles (y)
    const float* A = field ? Pv : Pu;
    float*       D = field ? DY : DX;

    int lane  = (int)(threadIdx.x & 31u);
    int lo    = lane & 15;
    int khalf = (lane >> 4) << 1;
    int row0  = tm * 16, col0 = tn * 16;

    v8f acc = {};
    for (int k0 = 0; k0 < KP; k0 += 4) {
        v2f a, b;
        const float* arow = A + (size_t)(row0 + lo) * KP + k0 + khalf;
        a.x = arow[0];
        a.y = arow[1];
        const float* brow = S + (size_t)(col0 + lo) * KP + k0 + khalf;
        b.x = brow[0];
        b.y = brow[1];
        acc = __builtin_amdgcn_wmma_f32_16x16x4_f32(
                  false, a, false, b, (short)0, acc, false, false);
    }
#pragma unroll
    for (int r = 0; r < 8; ++r) {
        int m = r + ((lane >> 4) << 3);
        D[(size_t)(row0 + m) * NN + col0 + lo] = acc[r];
    }
}

// ---------------------------------------------------------------------------
// remap: bilinear gather over all 384 channel planes.
// One thread per pixel; lanes walk x => coalesced stores, near-coalesced loads.
// DX/DY are stored transposed (U'[x,y] = u[y,x]); read accordingly.
// Output is streamed with non-temporal stores so the 100 MB write stream does
// not evict the (L2-resident, heavily reused) image planes.
// ---------------------------------------------------------------------------
__global__ void remap_kernel(const float* __restrict__ img,
                             const float* __restrict__ DX,
                             const float* __restrict__ DY,
                             float* __restrict__ out)
{
    int pix = blockIdx.x * blockDim.x + threadIdx.x;   // 0..65535
    if (pix >= NN * NN) return;
    int y = pix >> 8;
    int x = pix & (NN - 1);

    // displacement fields: read once, no reuse -> non-temporal loads
    float dx = __builtin_nontemporal_load(DX + x * NN + y);
    float dy = __builtin_nontemporal_load(DY + x * NN + y);

    float xn = fminf(fmaxf((float)x - dx, 0.0f), (float)(NN - 1));
    float yn = fminf(fmaxf((float)y - dy, 0.0f), (float)(NN - 1));

    float xff = floorf(xn), yff = floorf(yn);
    int xf = (int)xff;
    int yf = (int)yff;
    int xc = (int)ceilf(xn);
    int yc = (int)ceilf(yn);
    float xv = xn - xff;
    float yv = yn - yff;

    float w00 = (1.0f - yv) * (1.0f - xv);
    float w01 = (1.0f - yv) * xv;
    float w10 = yv * (1.0f - xv);
    float w11 = yv * xv;

    int o00 = yf * NN + xf;
    int o01 = yf * NN + xc;
    int o10 = yc * NN + xf;
    int o11 = yc * NN + xc;

    const size_t plane = (size_t)NN * NN;
#pragma unroll 4
    for (int c = 0; c < 384; ++c) {
        const float* p = img + (size_t)c * plane;
        float v = w00 * p[o00] + w01 * p[o01] + w10 * p[o10] + w11 * p[o11];
        __builtin_nontemporal_store(v, out + (size_t)c * plane + pix);
    }
}

// ---------------------------------------------------------------------------
extern "C" void kernel_launch(void* const* d_in, const int* in_sizes, int n_in,
                              void* d_out, int out_size, void* d_ws, size_t ws_size,
                              hipStream_t stream)
{
    (void)in_sizes; (void)n_in; (void)out_size; (void)ws_size;
    const float* img = (const float*)d_in[0];
    const float* cu  = (const float*)d_in[1];
    const float* cv  = (const float*)d_in[2];
    float* out = (float*)d_out;
    float* ws  = (float*)d_ws;

    // Deterministic temperature scale (host-side, pure math).
    const double PI = 3.14159265358979323846;
    double lg = log((double)CUTN + 1e-6);
    double T1 = 1.0 / (PI * (double)NN * (double)NN * lg);
    double T2 = 4.0 / (PI * PI * PI * (double)CUTN * (double)CUTN * lg);
    if (T2 < T1) T2 = T1;                 // ALPHA = 1.0
    double T = T1 + 0.5 * (T2 - T1);      // T_FRAC = 0.5
    float scale = (float)(sqrt(T) * (double)NN);

    // Workspace carve-up (floats): ~947 KB total.
    float* S  = ws;                       // 256*112
    float* Mu = S  + NN * KP;             // 112*112
    float* Mv = Mu + KP * KP;             // 112*112
    float* Pu = Mv + KP * KP;             // 256*112
    float* Pv = Pu + NN * KP;             // 256*112
    float* DX = Pv + NN * KP;             // 256*256
    float* DY = DX + NN * NN;             // 256*256

    int prep_elems = NN * KP + 2 * KP * KP;
    prep_kernel<<<(prep_elems + 255) / 256, 256, 0, stream>>>(cu, cv, S, Mu, Mv, scale);

    gemm_SM_kernel <<<2 * 16 * 7,  32, 0, stream>>>(S, Mu, Mv, Pu, Pv);
    gemm_PSt_kernel<<<2 * 16 * 16, 32, 0, stream>>>(S, Pu, Pv, DX, DY);

    remap_kernel<<<(NN * NN) / 256, 256, 0, stream>>>(img, DX, DY, out);
}